// OmniMambaCFDSolver_69423851372932
// MI455X (gfx1250) — compile-verified
//
#include <hip/hip_runtime.h>
#include <hip/hip_bf16.h>

// ---------------- problem constants ----------------
#define BB   4
#define INV  5
#define BCV  3
#define EE   96
#define NBLK 6
#define DSS  64
#define GG   16
#define LL   4096            // 16^3 voxels = sequence length
#define NN   (BB * LL)       // 16384 tokens
#define DIC  192             // 2*E
#define DTRC 6
#define XZC  384             // 2*DI
#define XPC  134             // DTR + 2*DS
#define XPCP 144             // padded to multiple of 16 for WMMA

typedef __attribute__((ext_vector_type(2))) float v2f;
typedef __attribute__((ext_vector_type(8))) float v8f;
typedef unsigned int u32x4 __attribute__((ext_vector_type(4)));
typedef unsigned int u32x8 __attribute__((ext_vector_type(8)));

__device__ __forceinline__ float silu_f(float v) { return v / (1.f + __expf(-v)); }

// =====================================================================
// Generic fp32 WMMA GEMM:  C[M][NC] = A[M][KDIM] * B (+bias)(+act)(+res)
// One wave per 16x16 tile; K stepped by 4 via V_WMMA_F32_16X16X4_F32.
// BTRANS=0: Bw stored [KDIM][NC] row-major. BTRANS=1: Bw stored [NC][KDIM].
// =====================================================================
template <int KDIM, int NC, int BTRANS, int HASBIAS, int ACT, int HASRES>
__global__ __launch_bounds__(32) void gemm_wmma_f32(
    const float* __restrict__ A, const float* __restrict__ Bw,
    const float* __restrict__ bias, const float* __restrict__ res,
    float* __restrict__ C)
{
    const int lane = threadIdx.x;       // 0..31, full wave => EXEC all ones
    const int g    = lane >> 4;         // half-wave select
    const int mr   = lane & 15;
    const int n0   = blockIdx.x * 16;
    const int m0   = blockIdx.y * 16;

    v8f acc = {0.f, 0.f, 0.f, 0.f, 0.f, 0.f, 0.f, 0.f};

    for (int k0 = 0; k0 < KDIM; k0 += 4) {
        const int kk = k0 + 2 * g;
        // A fragment: lane holds A[m][kk], A[m][kk+1]  (16x4, 2 VGPRs)
        v2f a;
        a.x = A[(size_t)(m0 + mr) * KDIM + kk];
        a.y = A[(size_t)(m0 + mr) * KDIM + kk + 1];
        // B fragment: lane holds B[kk][n], B[kk+1][n]  (4x16, 2 VGPRs)
        v2f b;
        if (BTRANS) {
            b.x = Bw[(size_t)(n0 + mr) * KDIM + kk];
            b.y = Bw[(size_t)(n0 + mr) * KDIM + kk + 1];
        } else {
            b.x = Bw[(size_t)kk * NC + n0 + mr];
            b.y = Bw[(size_t)(kk + 1) * NC + n0 + mr];
        }
        acc = __builtin_amdgcn_wmma_f32_16x16x4_f32(
            /*neg_a=*/false, a, /*neg_b=*/false, b,
            /*c_mod=*/(short)0, acc, /*reuse_a=*/false, /*reuse_b=*/false);
    }

    const int n = n0 + mr;
    const float bv = HASBIAS ? bias[n] : 0.f;
#pragma unroll
    for (int r = 0; r < 8; ++r) {
        const int m = m0 + r + 8 * g;   // C/D layout: VGPR r -> M = r (+8 for hi half)
        float v = acc[r] + bv;
        if (ACT == 1) v = silu_f(v);
        if (HASRES)  v += res[(size_t)m * NC + n];
        C[(size_t)m * NC + n] = v;
    }
}

// =====================================================================
// Front-end: fcat[n][0..95] = 3x3x3 conv(vles,Wf)+bf ; fcat[n][96..191] =
// Wb2 @ silu(Wb1 @ bc + bb1) + bb2 (pointwise).  One thread per (n,c).
// =====================================================================
__global__ void frontend_fcat_k(
    const float* __restrict__ vles, const float* __restrict__ bc,
    const float* __restrict__ Wf,  const float* __restrict__ bf,
    const float* __restrict__ Wb1, const float* __restrict__ bb1,
    const float* __restrict__ Wb2, const float* __restrict__ bb2,
    float* __restrict__ fcat)
{
    int idx = blockIdx.x * blockDim.x + threadIdx.x;
    if (idx >= NN * DIC) return;
    const int c = idx % DIC;
    const int n = idx / DIC;
    const int b = n / LL;
    const int l = n % LL;
    const int d = l >> 8, h = (l >> 4) & 15, w = l & 15;

    float out;
    if (c < EE) {
        float acc = bf[c];
        for (int ci = 0; ci < INV; ++ci)
            for (int kd = 0; kd < 3; ++kd) {
                const int dd = d + kd - 1; if ((unsigned)dd >= GG) continue;
                for (int kh = 0; kh < 3; ++kh) {
                    const int hh = h + kh - 1; if ((unsigned)hh >= GG) continue;
                    for (int kw = 0; kw < 3; ++kw) {
                        const int ww = w + kw - 1; if ((unsigned)ww >= GG) continue;
                        acc += Wf[(((c * INV + ci) * 3 + kd) * 3 + kh) * 3 + kw] *
                               vles[(((b * INV + ci) * GG + dd) * GG + hh) * GG + ww];
                    }
                }
            }
        out = acc;
    } else {
        const int e = c - EE;
        float acc = bb2[e];
        for (int j = 0; j < EE / 2; ++j) {
            float t = bb1[j];
            for (int ci = 0; ci < BCV; ++ci)
                t += Wb1[j * BCV + ci] * bc[(size_t)(b * BCV + ci) * LL + l];
            acc += Wb2[e * (EE / 2) + j] * silu_f(t);
        }
        out = acc;
    }
    fcat[(size_t)n * DIC + c] = out;
}

// =====================================================================
// LayerNorm over E=96 channels: one wave per token, 3 channels per lane,
// butterfly shuffle reductions (wave32).
// =====================================================================
__global__ __launch_bounds__(128) void layernorm_k(
    const float* __restrict__ X, const float* __restrict__ gam,
    const float* __restrict__ bet, float* __restrict__ Y)
{
    const int token = blockIdx.x * 4 + (threadIdx.x >> 5);
    const int lane  = threadIdx.x & 31;
    const float* row = X + (size_t)token * EE;
    float v0 = row[lane], v1 = row[lane + 32], v2 = row[lane + 64];
    float s = v0 + v1 + v2;
#pragma unroll
    for (int o = 16; o >= 1; o >>= 1) s += __shfl_xor(s, o, 32);
    const float mu = s * (1.f / 96.f);
    const float d0 = v0 - mu, d1 = v1 - mu, d2 = v2 - mu;
    float q = d0 * d0 + d1 * d1 + d2 * d2;
#pragma unroll
    for (int o = 16; o >= 1; o >>= 1) q += __shfl_xor(q, o, 32);
    const float rinv = rsqrtf(q * (1.f / 96.f) + 1e-5f);
    float* out = Y + (size_t)token * EE;
    out[lane]      = d0 * rinv * gam[lane]      + bet[lane];
    out[lane + 32] = d1 * rinv * gam[lane + 32] + bet[lane + 32];
    out[lane + 64] = d2 * rinv * gam[lane + 64] + bet[lane + 64];
}

// ============ depthwise causal conv1d (K=4) + SiLU over xz[..., :192] ====
__global__ void convsilu_k(const float* __restrict__ xz,
                           const float* __restrict__ cw,
                           const float* __restrict__ cb,
                           float* __restrict__ xi)
{
    int idx = blockIdx.x * blockDim.x + threadIdx.x;
    if (idx >= NN * DIC) return;
    const int d = idx % DIC;
    const int n = idx / DIC;
    const int b = n / LL, l = n % LL;
    float acc = cb[d];
#pragma unroll
    for (int k = 0; k < 4; ++k) {
        const int lp = l + k - 3;
        const float v = (lp >= 0) ? xz[(size_t)(b * LL + lp) * XZC + d] : 0.f;
        acc += cw[d * 4 + k] * v;
    }
    xi[(size_t)n * DIC + d] = silu_f(acc);
}

// ============ pad Wx (192x134) -> (192x144) with zeros ============
__global__ void padwx_k(const float* __restrict__ Wx, float* __restrict__ Wxp)
{
    int idx = blockIdx.x * blockDim.x + threadIdx.x;
    if (idx >= DIC * XPCP) return;
    const int k = idx / XPCP, n = idx % XPCP;
    Wxp[idx] = (n < XPC) ? Wx[k * XPC + n] : 0.f;
}

// ============ delta = softplus(dt @ Wdt + bdt) ============
__global__ void delta_k(const float* __restrict__ xp,
                        const float* __restrict__ Wdt,
                        const float* __restrict__ bdt,
                        float* __restrict__ de)
{
    int idx = blockIdx.x * blockDim.x + threadIdx.x;
    if (idx >= NN * DIC) return;
    const int d = idx % DIC;
    const int n = idx / DIC;
    float acc = bdt[d];
#pragma unroll
    for (int r = 0; r < DTRC; ++r)
        acc += xp[(size_t)n * XPCP + r] * Wdt[r * DIC + d];
    de[(size_t)n * DIC + d] = (acc > 20.f) ? acc : log1pf(__expf(acc));
}

// =====================================================================
// Selective scan with TDM double-buffered staging.
// One wave per (b,d) recurrence; 2 state elems per lane (DS=64, wave32).
// 8 waves share a B/C tile staged by the Tensor Data Mover: xp columns
// 6..133 are 128 contiguous floats per row, so each 32-step chunk is one
// 2D TDM tile (32 rows x 512B, row stride 576B) -> one tensor_load_to_lds
// per chunk, issued by wave 0 for chunk c+1 while all waves compute chunk c.
// Completion via TENSORcnt (in-order per wave): wait <=1 after issuing next.
// =====================================================================
#define DG 8    // d-channels (waves) per workgroup
#define CH 32   // timesteps per staged chunk

__device__ __forceinline__ void tdm_load_tile(unsigned lds_byte_addr,
                                              const float* gptr)
{
    const unsigned long long ga = (unsigned long long)(uintptr_t)gptr;
    u32x4 g0;
    g0[0] = 1u;                                   // count=1, user descriptor
    g0[1] = lds_byte_addr;                        // LDS dest (bytes)
    g0[2] = (unsigned)(ga & 0xffffffffu);         // global_addr[31:0]
    g0[3] = (unsigned)((ga >> 32) & 0x01ffffffu)  // global_addr[56:32]
          | (2u << 30);                           // type=2 ("image")
    u32x8 g1;
    g1[0] = 2u << 16;        // wg_mask=0, data_size=2 (4 bytes), no flags
    g1[1] = 128u << 16;      // tensor_dim0[15:0]=128   (bits 79:48)
    g1[2] = 32u << 16;       // tensor_dim0 hi=0 | tensor_dim1[15:0]=32
    g1[3] = 128u << 16;      // tensor_dim1 hi=0 | tile_dim0=128
    g1[4] = 32u;             // tile_dim1=32, tile_dim2=0 (2D tile)
    g1[5] = 144u;            // tensor_dim0_stride low 32 (XPCP elements)
    g1[6] = 0u;              // stride0 hi | tensor_dim1_stride lo (unused)
    g1[7] = 0u;
#ifdef __HIP_DEVICE_COMPILE__
    asm volatile("tensor_load_to_lds %0, %1" :: "s"(g0), "s"(g1) : "memory");
#endif
}

__global__ __launch_bounds__(256) void scan_k(
    const float* __restrict__ xp,  const float* __restrict__ xi,
    const float* __restrict__ de,  const float* __restrict__ A_log,
    float* __restrict__ y)
{
    __shared__ float sBC[2][CH * 128];   // double-buffered TDM tile: [st][B(64)|C(64)]
    __shared__ float sU[DG * CH];
    __shared__ float sD[DG * CH];

    const int b  = blockIdx.x / (DIC / DG);
    const int dg = blockIdx.x % (DIC / DG);
    const int t  = threadIdx.x;
    const int w  = t >> 5, lane = t & 31;
    const int d  = dg * DG + w;

    const float a0 = -__expf(A_log[d * DSS + lane]);
    const float a1 = -__expf(A_log[d * DSS + lane + 32]);
    float h0 = 0.f, h1 = 0.f;
    const size_t nbase = (size_t)b * LL;
    const unsigned ldsbase = (unsigned)(uintptr_t)&sBC[0][0];

    // prologue: DMA chunk 0 into buffer 0 (wave 0 only; TDM is a wave op)
    if (w == 0)
        tdm_load_tile(ldsbase, xp + (nbase) * XPCP + DTRC);

    const int NCHUNK = LL / CH;
    for (int c = 0; c < NCHUNK; ++c) {
        const int l0  = c * CH;
        const int buf = c & 1;

        // per-thread u/delta for this chunk (DG*CH == 256 == blockDim)
        const int wv = t >> 5, st0 = t & 31;
        const float ureg = xi[(nbase + l0 + st0) * DIC + dg * DG + wv];
        const float dreg = de[(nbase + l0 + st0) * DIC + dg * DG + wv];

        if (w == 0) {
            if (c + 1 < NCHUNK) {
                // DMA next chunk into the other buffer while we compute
                tdm_load_tile(ldsbase + (unsigned)((buf ^ 1) * CH * 128 * 4),
                              xp + (nbase + l0 + CH) * XPCP + DTRC);
                __builtin_amdgcn_s_wait_tensorcnt(1);  // chunk c done (in-order)
            } else {
                __builtin_amdgcn_s_wait_tensorcnt(0);
            }
        }
        sU[t] = ureg;
        sD[t] = dreg;
        __syncthreads();    // tile buf + sU/sD visible to all waves

        // prefetch next chunk's u/delta rows into cache (global_prefetch_b8)
        if (c + 1 < NCHUNK) {
            __builtin_prefetch(xi + (nbase + l0 + CH + st0) * DIC + dg * DG, 0, 0);
            __builtin_prefetch(de + (nbase + l0 + CH + st0) * DIC + dg * DG, 0, 0);
        }

        const float* tile = &sBC[buf][0];
        for (int st = 0; st < CH; ++st) {
            const float dl = sD[w * CH + st];        // LDS broadcast
            const float u  = sU[w * CH + st];
            const float du = dl * u;
            const float bm0 = tile[st * 128 + lane];
            const float bm1 = tile[st * 128 + lane + 32];
            const float cm0 = tile[st * 128 + 64 + lane];
            const float cm1 = tile[st * 128 + 96 + lane];
            h0 = __expf(dl * a0) * h0 + du * bm0;
            h1 = __expf(dl * a1) * h1 + du * bm1;
            float p = h0 * cm0 + h1 * cm1;
#pragma unroll
            for (int o = 16; o >= 1; o >>= 1) p += __shfl_xor(p, o, 32);
            if (lane == 0) y[(nbase + l0 + st) * DIC + d] = p;
        }
        __syncthreads();    // all waves done with buf & sU/sD before overwrite
    }
}

// ============ y = (ys + xi*Dp) * silu(z) ============
__global__ void ycomb_k(const float* __restrict__ xi,
                        const float* __restrict__ xz,
                        const float* __restrict__ Dp,
                        float* __restrict__ y)
{
    int idx = blockIdx.x * blockDim.x + threadIdx.x;
    if (idx >= NN * DIC) return;
    const int d = idx % DIC;
    const int n = idx / DIC;
    const float z = xz[(size_t)n * XZC + DIC + d];
    y[idx] = (y[idx] + xi[idx] * Dp[d]) * silu_f(z);
}

// ============ final 3x3x3 conv (192 -> 5) + residual with vles ============
__global__ void finalconv_k(const float* __restrict__ dbuf,
                            const float* __restrict__ Wd2,
                            const float* __restrict__ bd2,
                            const float* __restrict__ vles,
                            float* __restrict__ out)
{
    int idx = blockIdx.x * blockDim.x + threadIdx.x;
    if (idx >= BB * INV * LL) return;
    const int l  = idx % LL;
    const int vo = (idx / LL) % INV;
    const int b  = idx / (LL * INV);
    const int d = l >> 8, h = (l >> 4) & 15, w = l & 15;

    float acc = bd2[vo];
    for (int kd = 0; kd < 3; ++kd) {
        const int dd = d + kd - 1; if ((unsigned)dd >= GG) continue;
        for (int kh = 0; kh < 3; ++kh) {
            const int hh = h + kh - 1; if ((unsigned)hh >= GG) continue;
            for (int kw = 0; kw < 3; ++kw) {
                const int ww = w + kw - 1; if ((unsigned)ww >= GG) continue;
                const float* drow = dbuf + ((size_t)b * LL + (dd * 256 + hh * 16 + ww)) * DIC;
                float s = 0.f;
                for (int c = 0; c < DIC; ++c)
                    s += drow[c] * Wd2[(((vo * DIC + c) * 3 + kd) * 3 + kh) * 3 + kw];
                acc += s;
            }
        }
    }
    out[idx] = vles[idx] + acc;
}

// =====================================================================
// Host-side orchestration
// =====================================================================
extern "C" void kernel_launch(void* const* d_in, const int* in_sizes, int n_in,
                              void* d_out, int out_size, void* d_ws, size_t ws_size,
                              hipStream_t stream)
{
    (void)in_sizes; (void)n_in; (void)out_size; (void)ws_size;
    const float* vles = (const float*)d_in[0];
    const float* bc   = (const float*)d_in[1];
    const float* Wf   = (const float*)d_in[2];
    const float* bf   = (const float*)d_in[3];
    const float* Wb1  = (const float*)d_in[4];
    const float* bb1  = (const float*)d_in[5];
    const float* Wb2  = (const float*)d_in[6];
    const float* bb2  = (const float*)d_in[7];
    const float* Wfu  = (const float*)d_in[8];
    const float* bfu  = (const float*)d_in[9];
    const float* ln_g = (const float*)d_in[10];
    const float* ln_b = (const float*)d_in[11];
    const float* Win  = (const float*)d_in[12];
    const float* cw   = (const float*)d_in[13];
    const float* cb   = (const float*)d_in[14];
    const float* Wx   = (const float*)d_in[15];
    const float* Wdt  = (const float*)d_in[16];
    const float* bdt  = (const float*)d_in[17];
    const float* A_log= (const float*)d_in[18];
    const float* Dp   = (const float*)d_in[19];
    const float* Wout = (const float*)d_in[20];
    const float* Wd1  = (const float*)d_in[21];
    const float* bd1  = (const float*)d_in[22];
    const float* Wd2  = (const float*)d_in[23];
    const float* bd2  = (const float*)d_in[24];

    // workspace carve-out (floats)
    float* ws = (float*)d_ws;
    size_t o = 0;
    float* x0  = ws + o; o += (size_t)NN * EE;
    float* x1  = ws + o; o += (size_t)NN * EE;
    float* tn  = ws + o; o += (size_t)NN * EE;
    float* xz  = ws + o; o += (size_t)NN * XZC;
    float* xi  = ws + o; o += (size_t)NN * DIC;   // fcat / xi / decoder dbuf
    float* xp  = ws + o; o += (size_t)NN * XPCP;
    float* de  = ws + o; o += (size_t)NN * DIC;
    float* yb  = ws + o; o += (size_t)NN * DIC;
    float* Wxp = ws + o; o += (size_t)DIC * XPCP;

    const int TPB = 256;
    const int nElemND = NN * DIC;

    // ---- front-end: fcat then fuse GEMM (96 <- 192) ----
    frontend_fcat_k<<<(nElemND + TPB - 1) / TPB, TPB, 0, stream>>>(
        vles, bc, Wf, bf, Wb1, bb1, Wb2, bb2, xi);
    gemm_wmma_f32<DIC, EE, 1, 1, 0, 0><<<dim3(EE / 16, NN / 16), 32, 0, stream>>>(
        xi, Wfu, bfu, nullptr, x0);

    float* xc = x0;
    float* xn = x1;

    // ---- 6 Mamba blocks ----
    for (int i = 0; i < NBLK; ++i) {
        layernorm_k<<<NN / 4, 128, 0, stream>>>(xc, ln_g + i * EE, ln_b + i * EE, tn);

        gemm_wmma_f32<EE, XZC, 0, 0, 0, 0><<<dim3(XZC / 16, NN / 16), 32, 0, stream>>>(
            tn, Win + (size_t)i * EE * XZC, nullptr, nullptr, xz);

        convsilu_k<<<(nElemND + TPB - 1) / TPB, TPB, 0, stream>>>(
            xz, cw + i * DIC * 4, cb + i * DIC, xi);

        padwx_k<<<(DIC * XPCP + TPB - 1) / TPB, TPB, 0, stream>>>(
            Wx + (size_t)i * DIC * XPC, Wxp);

        gemm_wmma_f32<DIC, XPCP, 0, 0, 0, 0><<<dim3(XPCP / 16, NN / 16), 32, 0, stream>>>(
            xi, Wxp, nullptr, nullptr, xp);

        delta_k<<<(nElemND + TPB - 1) / TPB, TPB, 0, stream>>>(
            xp, Wdt + i * DTRC * DIC, bdt + i * DIC, de);

        scan_k<<<BB * (DIC / DG), 256, 0, stream>>>(
            xp, xi, de, A_log + (size_t)i * DIC * DSS, yb);

        ycomb_k<<<(nElemND + TPB - 1) / TPB, TPB, 0, stream>>>(
            xi, xz, Dp + i * DIC, yb);

        gemm_wmma_f32<DIC, EE, 0, 0, 0, 1><<<dim3(EE / 16, NN / 16), 32, 0, stream>>>(
            yb, Wout + (size_t)i * DIC * EE, nullptr, xc, xn);

        float* tmp = xc; xc = xn; xn = tmp;
    }

    // ---- decoder: d = silu(x @ Wd1^T + bd1); out = vles + conv3(d, Wd2) ----
    gemm_wmma_f32<EE, DIC, 1, 1, 1, 0><<<dim3(DIC / 16, NN / 16), 32, 0, stream>>>(
        xc, Wd1, bd1, nullptr, xi);

    finalconv_k<<<(BB * INV * LL + TPB - 1) / TPB, TPB, 0, stream>>>(
        xi, Wd2, bd2, vles, (float*)d_out);
}